// ProteinFeatures_42176578846967
// MI455X (gfx1250) — compile-verified
//
#include <hip/hip_runtime.h>
#include <hip/hip_bf16.h>

#define BB 2
#define NN 2048
#define TOPK 48
#define NRBF 16
#define MAXREL 32
#define POSIN 66           // 2*MAXREL + 2
#define EF 128             // EDGE_FEATURES
#define KRBF 400           // 25 * 16  (rbf part of EDGE_IN)
#define FSTRIDE 402        // padded LDS row stride (even -> 8B aligned float2)
#define BIGF 3.0e38f

typedef __attribute__((ext_vector_type(2))) float v2f;
typedef __attribute__((ext_vector_type(8))) float v8f;

// atom codes: 0=N 1=Ca 2=C 3=O 4=Cb ; entry 0 is the D_neighbors (special) slot
__device__ __constant__ int PA[25] = {1, 0,2,3,4, 1,1,1,1, 0,0,0, 4,4, 3, 0,2,3,4, 2,3,4, 2,3, 2};
__device__ __constant__ int PB[25] = {1, 0,2,3,4, 0,2,3,4, 2,3,4, 2,3, 2, 1,1,1,1, 0,0,0, 4,4, 3};

// ---------------------------------------------------------------- stage 1a
__global__ void build_coords_kernel(const float* __restrict__ X,
                                    float* __restrict__ coords) {
  int bi = blockIdx.x * blockDim.x + threadIdx.x;
  if (bi >= BB * NN) return;
  const float* x = X + (size_t)bi * 12;
  float nx = x[0], ny = x[1], nz = x[2];
  float cax = x[3], cay = x[4], caz = x[5];
  float cx = x[6], cy = x[7], cz = x[8];
  float ox = x[9], oy = x[10], oz = x[11];
  float bx = cax - nx, by = cay - ny, bz = caz - nz;     // b = Ca - N
  float ccx = cx - cax, ccy = cy - cay, ccz = cz - caz;  // c = C - Ca
  float ax = by * ccz - bz * ccy;                        // a = b x c
  float ay = bz * ccx - bx * ccz;
  float az = bx * ccy - by * ccx;
  float cbx = -0.58273431f * ax + 0.56802827f * bx - 0.54067466f * ccx + cax;
  float cby = -0.58273431f * ay + 0.56802827f * by - 0.54067466f * ccy + cay;
  float cbz = -0.58273431f * az + 0.56802827f * bz - 0.54067466f * ccz + caz;
  float* o = coords + (size_t)bi * 15;
  o[0] = nx;  o[1] = ny;  o[2] = nz;
  o[3] = cax; o[4] = cay; o[5] = caz;
  o[6] = cx;  o[7] = cy;  o[8] = cz;
  o[9] = ox;  o[10] = oy; o[11] = oz;
  o[12] = cbx; o[13] = cby; o[14] = cbz;
}

// ---------------------------------------------------------------- stage 1b
// Wcomb[d][n] = sum_p (W_pos[d][p]+b_pos[p]) * W_edge[p][n] + b_edge[n]
__global__ void wcomb_kernel(const float* __restrict__ W_pos,
                             const float* __restrict__ b_pos,
                             const float* __restrict__ W_edge,
                             const float* __restrict__ b_edge,
                             float* __restrict__ Wcomb) {
  int t = blockIdx.x * blockDim.x + threadIdx.x;
  if (t >= POSIN * EF) return;
  int d = t / EF, n = t % EF;
  float s = b_edge[n];
  for (int p = 0; p < 16; ++p)
    s += (W_pos[d * 16 + p] + b_pos[p]) * W_edge[p * EF + n];
  Wcomb[t] = s;
}

// ---------------------------------------------------------------- stage 1c
// Pack W_edge rows 16..415 into (k/2, n)-major float2 pairs so the WMMA
// B-fragment is a single 8-byte global load: Wpack[kk/2][n] = {W[kk][n], W[kk+1][n]}
__global__ void wpack_kernel(const float* __restrict__ W_edge,
                             v2f* __restrict__ Wpack) {
  int t = blockIdx.x * blockDim.x + threadIdx.x;
  if (t >= (KRBF / 2) * EF) return;
  int kp = t / EF, n = t % EF;
  const float* Wr = W_edge + 16 * EF;
  v2f v;
  v.x = Wr[(2 * kp) * EF + n];
  v.y = Wr[(2 * kp + 1) * EF + n];
  Wpack[t] = v;
}

// ---------------------------------------------------------------- stage 2
__global__ __launch_bounds__(256)
void topk_kernel(const float* __restrict__ coords,
                 const float* __restrict__ mask,
                 const int* __restrict__ Ridx,
                 const int* __restrict__ chains,
                 float* __restrict__ Dn,
                 int* __restrict__ dcode,
                 int* __restrict__ Eidx) {
  __shared__ float dist[NN];
  __shared__ float rv[256];
  __shared__ int ri[256];
  int tid = threadIdx.x;
  int bi = blockIdx.x;
  int b = bi / NN;
  const float* ci = coords + (size_t)bi * 15 + 3;  // Ca of i
  float cax = ci[0], cay = ci[1], caz = ci[2];
  float mi = mask[bi];

  float lmax = 0.0f;
  for (int j = tid; j < NN; j += 256) {
    const float* cj = coords + (size_t)(b * NN + j) * 15 + 3;
    float dx = cax - cj[0], dy = cay - cj[1], dz = caz - cj[2];
    float d = sqrtf(dx * dx + dy * dy + dz * dz + 1e-6f);
    float m2 = mi * mask[b * NN + j];
    float Dv = m2 * d;
    dist[j] = Dv;
    lmax = fmaxf(lmax, Dv);
  }
  rv[tid] = lmax;
  __syncthreads();
  for (int s = 128; s > 0; s >>= 1) {
    if (tid < s) rv[tid] = fmaxf(rv[tid], rv[tid + s]);
    __syncthreads();
  }
  float rowmax = rv[0];
  __syncthreads();
  for (int j = tid; j < NN; j += 256) {
    float m2 = mi * mask[b * NN + j];
    dist[j] = dist[j] + (1.0f - m2) * rowmax;
  }
  __syncthreads();

  for (int k = 0; k < TOPK; ++k) {
    float bv = BIGF;
    int bidx = 0x7FFFFFFF;
    for (int j = tid; j < NN; j += 256) {
      float v = dist[j];
      if (v < bv) { bv = v; bidx = j; }
    }
    rv[tid] = bv; ri[tid] = bidx;
    __syncthreads();
    for (int s = 128; s > 0; s >>= 1) {
      if (tid < s) {
        float v2 = rv[tid + s]; int i2 = ri[tid + s];
        if (v2 < rv[tid] || (v2 == rv[tid] && i2 < ri[tid])) { rv[tid] = v2; ri[tid] = i2; }
      }
      __syncthreads();
    }
    if (tid == 0) {
      int j = ri[0];
      Eidx[bi * TOPK + k] = j;
      Dn[bi * TOPK + k] = rv[0];
      int off = Ridx[bi] - Ridx[b * NN + j];
      int eq = (chains[bi] == chains[b * NN + j]) ? 1 : 0;
      int d = off + MAXREL;
      d = d < 0 ? 0 : (d > 2 * MAXREL ? 2 * MAXREL : d);
      dcode[bi * TOPK + k] = eq ? d : (2 * MAXREL + 1);
      dist[j] = BIGF;
    }
    __syncthreads();
  }
}

// ---------------------------------------------------------------- stage 3
// One WG per residue: build 48x400 RBF features in LDS, then
// f32 WMMA GEMM against packed W_edge[16:416,:] + Wcomb[dcode] lookup epilogue.
__global__ __launch_bounds__(256)
void edge_kernel(const float* __restrict__ coords,
                 const v2f* __restrict__ Wpack,
                 const float* __restrict__ Wcomb,
                 const float* __restrict__ Dn,
                 const int* __restrict__ dcode,
                 const int* __restrict__ Eidx,
                 float* __restrict__ Eout) {
  __shared__ __align__(16) float F[TOPK * FSTRIDE];
  __shared__ int s_idx[TOPK];
  __shared__ int s_dc[TOPK];
  __shared__ float s_dn[TOPK];
  __shared__ float s_ci[15];
  __shared__ float s_cj[TOPK * 15];

  int tid = threadIdx.x;
  int bi = blockIdx.x;
  int b = bi / NN;

  if (tid < TOPK) {
    s_idx[tid] = Eidx[bi * TOPK + tid];
    s_dc[tid]  = dcode[bi * TOPK + tid];
    s_dn[tid]  = Dn[bi * TOPK + tid];
  }
  if (tid >= 64 && tid < 79) s_ci[tid - 64] = coords[(size_t)bi * 15 + (tid - 64)];
  __syncthreads();
  for (int t = tid; t < TOPK * 15; t += 256) {
    int e = t / 15, r = t % 15;
    s_cj[t] = coords[(size_t)(b * NN + s_idx[e]) * 15 + r];
  }
  __syncthreads();

  // feature fill: 48 edges x 25 pair-distances x 16 RBFs
  for (int t = tid; t < TOPK * 25; t += 256) {
    int e = t / 25, p = t % 25;
    float d;
    if (p == 0) {
      d = s_dn[e];
    } else {
      int a = PA[p], bb = PB[p];
      float dx = s_ci[a * 3 + 0] - s_cj[e * 15 + bb * 3 + 0];
      float dy = s_ci[a * 3 + 1] - s_cj[e * 15 + bb * 3 + 1];
      float dz = s_ci[a * 3 + 2] - s_cj[e * 15 + bb * 3 + 2];
      d = sqrtf(dx * dx + dy * dy + dz * dz + 1e-6f);
    }
#pragma unroll
    for (int r = 0; r < NRBF; ++r) {
      float mu = 2.0f + (20.0f / 15.0f) * (float)r;
      float z = (d - mu) * (1.0f / 1.25f);
      F[e * FSTRIDE + p * 16 + r] = __expf(-z * z);
    }
  }
  __syncthreads();

  // GEMM: 8 waves; wave w -> output columns [16w, 16w+16), 3 row-tiles of 16
  int wave = tid >> 5;
  int lane = tid & 31;
  int ln = lane & 15;
  int klo = (lane >> 4) << 1;  // 0 for lanes 0-15, 2 for lanes 16-31
  int n0 = wave * 16;

  v8f acc0 = {}, acc1 = {}, acc2 = {};
  for (int kk = 0; kk < KRBF; kk += 4) {
    int khalf = (kk + klo) >> 1;
    v2f bf = Wpack[khalf * EF + n0 + ln];                 // one global_load_b64
    v2f a0 = *reinterpret_cast<const v2f*>(&F[(0  + ln) * FSTRIDE + kk + klo]);
    v2f a1 = *reinterpret_cast<const v2f*>(&F[(16 + ln) * FSTRIDE + kk + klo]);
    v2f a2 = *reinterpret_cast<const v2f*>(&F[(32 + ln) * FSTRIDE + kk + klo]);
    acc0 = __builtin_amdgcn_wmma_f32_16x16x4_f32(false, a0, false, bf, (short)0, acc0, false, false);
    acc1 = __builtin_amdgcn_wmma_f32_16x16x4_f32(false, a1, false, bf, (short)0, acc1, false, false);
    acc2 = __builtin_amdgcn_wmma_f32_16x16x4_f32(false, a2, false, bf, (short)0, acc2, false, false);
  }

  // epilogue: D layout -> lane<16: rows m0+r ; lane>=16: rows m0+8+r ; col n0+(lane&15)
  int row_off = (lane >> 4) << 3;
  int col = n0 + ln;
  size_t base = (size_t)bi * TOPK * EF;
#pragma unroll
  for (int r = 0; r < 8; ++r) {
    int r0 = r + row_off;
    int r1 = 16 + r + row_off;
    int r2 = 32 + r + row_off;
    Eout[base + (size_t)r0 * EF + col] = acc0[r] + Wcomb[s_dc[r0] * EF + col];
    Eout[base + (size_t)r1 * EF + col] = acc1[r] + Wcomb[s_dc[r1] * EF + col];
    Eout[base + (size_t)r2 * EF + col] = acc2[r] + Wcomb[s_dc[r2] * EF + col];
  }
}

// ----------------------------------------------------------------
extern "C" void kernel_launch(void* const* d_in, const int* in_sizes, int n_in,
                              void* d_out, int out_size, void* d_ws, size_t ws_size,
                              hipStream_t stream) {
  const float* X      = (const float*)d_in[0];
  const float* mask   = (const float*)d_in[1];
  const int*   Ridx   = (const int*)d_in[2];
  const int*   chains = (const int*)d_in[3];
  const float* W_pos  = (const float*)d_in[4];
  const float* b_pos  = (const float*)d_in[5];
  const float* W_edge = (const float*)d_in[6];
  const float* b_edge = (const float*)d_in[7];

  float* Eout = (float*)d_out;
  int*   EidxOut = (int*)(Eout + (size_t)BB * NN * TOPK * EF);  // tuple tail

  // workspace layout (all offsets 8B-aligned)
  float* coords = (float*)d_ws;                              // B*N*5*3   = 61440 f
  float* Dn     = coords + (size_t)BB * NN * 15;             // B*N*48    = 196608 f
  int*   dcode  = (int*)(Dn + (size_t)BB * NN * TOPK);       // B*N*48    = 196608 i
  float* Wcomb  = (float*)(dcode + (size_t)BB * NN * TOPK);  // 66*128    = 8448 f
  v2f*   Wpack  = (v2f*)(Wcomb + (size_t)POSIN * EF);        // 200*128 float2 = 204.8KB

  build_coords_kernel<<<(BB * NN + 255) / 256, 256, 0, stream>>>(X, coords);
  wcomb_kernel<<<(POSIN * EF + 255) / 256, 256, 0, stream>>>(W_pos, b_pos, W_edge, b_edge, Wcomb);
  wpack_kernel<<<((KRBF / 2) * EF + 255) / 256, 256, 0, stream>>>(W_edge, Wpack);
  topk_kernel<<<BB * NN, 256, 0, stream>>>(coords, mask, Ridx, chains, Dn, dcode, EidxOut);
  edge_kernel<<<BB * NN, 256, 0, stream>>>(coords, Wpack, Wcomb, Dn, dcode, EidxOut, Eout);
}